// MixA_Module_22935125361066
// MI455X (gfx1250) — compile-verified
//
#include <hip/hip_runtime.h>

// Problem constants (from reference): x:(4,64,32,64,64) f32, skin:(4,32,64,64) f32
#define BB 4
#define CC 64
#define TT 32
#define SPATIAL 4096                    // W*H = 64*64
#define CH_STRIDE (TT * SPATIAL)        // stride between channels = 131072 elems
#define FRAMES (BB * TT)                // 128
#define OUT_ELEMS ((size_t)BB * CC * TT * SPATIAL)   // 33,554,432

typedef float v2f __attribute__((ext_vector_type(2)));
typedef float v4f __attribute__((ext_vector_type(4)));
typedef float v8f __attribute__((ext_vector_type(8)));

__device__ __forceinline__ float fast_sigmoid(float z) {
    return 1.0f / (1.0f + __expf(-z));
}

// ---------------------------------------------------------------------------
// Kernel A: one workgroup per (b,t) frame. Computes
//   fusion = sigmoid(mean_c x) + sigmoid(max_c x) + skin
//   attn   = softmax over the 4096 spatial positions (shift by constant 3.0,
//            exact since fusion in (0,3))
// Channel SUM is done on the matrix unit: V_WMMA_F32_16X16X4_F32 with A=ones,
// accumulating 4 channels x 16 positions per instruction, chained 16x.
// Channel MAX rides along in VALU + one cross-half-wave shuffle.
// ---------------------------------------------------------------------------
__global__ __launch_bounds__(1024) void attn_frame_kernel(
    const float* __restrict__ x,
    const float* __restrict__ skin,
    float* __restrict__ attn_out)
{
    __shared__ float e_lds[SPATIAL];   // 16 KB of the 320 KB WGP LDS
    __shared__ float s_sum;

    const int f    = blockIdx.x;       // frame id = b*T + t
    const int b    = f >> 5;           // T = 32
    const int t    = f & 31;
    const int tid  = threadIdx.x;
    const int wave = tid >> 5;         // 32 waves (wave32)
    const int lane = tid & 31;
    const int half = lane >> 4;        // 0: channels {cb,cb+1}, 1: {cb+2,cb+3}
    const int ln   = lane & 15;        // position within 16-wide WMMA column set

    if (tid == 0) s_sum = 0.0f;
    __syncthreads();

    // x(b, c, t, p) = b*C*T*SPATIAL + c*CH_STRIDE + t*SPATIAL + p
    const size_t xbase = (size_t)b * CC * CH_STRIDE + (size_t)t * SPATIAL;

    v2f ones; ones.x = 1.0f; ones.y = 1.0f;   // A-matrix: 16x4 of ones

    float partial = 0.0f;                     // exp-sum partial (half==0 lanes)

    // Each wave covers 128 positions in 8 chunks of 16.
    #pragma unroll 1
    for (int k = 0; k < 8; ++k) {
        const int pos = (wave * 8 + k) * 16 + ln;
        const float* xp = x + xbase + (size_t)pos + (size_t)(2 * half) * CH_STRIDE;

        v8f acc = {};                         // f32 accumulator: column sums
        float m = -3.402823466e+38f;

        #pragma unroll
        for (int cb = 0; cb < CC; cb += 4) {
            const float v0 = xp[(size_t)cb * CH_STRIDE];
            const float v1 = xp[(size_t)(cb + 1) * CH_STRIDE];
            m = fmaxf(m, fmaxf(v0, v1));
            v2f bv; bv.x = v0; bv.y = v1;
            // D[:,n] += sum_k B[k][n]  (A = ones) -> channel partial sums
            acc = __builtin_amdgcn_wmma_f32_16x16x4_f32(
                false, ones, false, bv, (short)0, acc, false, false);
        }

        // finish 64-channel max: combine the two half-waves feeding column ln
        m = fmaxf(m, __shfl_xor(m, 16, 32));

        // acc[0] in lane L = full 64-channel sum for position (pbase + ln)
        const float avg = acc[0] * (1.0f / 64.0f);
        const float fus = fast_sigmoid(avg) + fast_sigmoid(m)
                        + skin[(size_t)f * SPATIAL + pos];
        const float e = __expf(fus - 3.0f);   // stable: fus in (0,3)

        if (half == 0) {                       // each position exactly once
            e_lds[pos] = e;
            partial += e;
        }
    }

    // wave reduction (upper-half lanes hold 0), then one LDS float atomic/wave
    #pragma unroll
    for (int off = 16; off >= 1; off >>= 1)
        partial += __shfl_xor(partial, off, 32);
    if (lane == 0) atomicAdd(&s_sum, partial);
    __syncthreads();

    const float inv = 1.0f / s_sum;
    for (int p = tid; p < SPATIAL; p += 1024)
        attn_out[(size_t)f * SPATIAL + p] = e_lds[p] * inv;
}

// ---------------------------------------------------------------------------
// Kernel B: out = x * attn, pure streaming at HBM rate. x was just read by
// kernel A and (134 MB) fits in the 192 MB L2, so these loads should hit L2.
// Non-temporal loads (last use of x) and non-temporal stores (out is never
// re-read) keep the 134 MB of stores from thrashing L2.
// ---------------------------------------------------------------------------
__global__ __launch_bounds__(256) void apply_attn_kernel(
    const float* __restrict__ x,
    const float* __restrict__ attn,
    float* __restrict__ out)
{
    const size_t n4 = OUT_ELEMS / 4;         // float4 elements: 8,388,608
    const size_t stride = (size_t)gridDim.x * blockDim.x;
    for (size_t i = (size_t)blockIdx.x * blockDim.x + threadIdx.x;
         i < n4; i += stride) {
        const size_t e0 = i * 4;
        const int p  = (int)(e0 & (SPATIAL - 1));
        const int tt = (int)((e0 >> 12) & 31);
        const int bc = (int)(e0 >> 17);
        const int bb = bc >> 6;
        const size_t aidx = ((size_t)(bb * 32 + tt) << 12) + (size_t)p;

        v4f xv = __builtin_nontemporal_load((const v4f*)x + i);
        v4f av = *(const v4f*)(attn + aidx);   // RT: reused 64x across channels
        v4f o  = xv * av;
        __builtin_nontemporal_store(o, (v4f*)out + i);
    }
}

extern "C" void kernel_launch(void* const* d_in, const int* in_sizes, int n_in,
                              void* d_out, int out_size, void* d_ws, size_t ws_size,
                              hipStream_t stream)
{
    const float* x    = (const float*)d_in[0];
    const float* skin = (const float*)d_in[1];
    float* out  = (float*)d_out;
    float* attn = out + OUT_ELEMS;   // tuple output: (output, attn) concatenated

    attn_frame_kernel<<<FRAMES, 1024, 0, stream>>>(x, skin, attn);
    apply_attn_kernel<<<8192, 256, 0, stream>>>(x, attn, out);
}